// Loss_57458072486030
// MI455X (gfx1250) — compile-verified
//
#include <hip/hip_runtime.h>
#include <math.h>

typedef __attribute__((ext_vector_type(2))) float v2f;
typedef __attribute__((ext_vector_type(8))) float v8f;

#define BATCH  8
#define NPTS   2048
#define NPOINT 102          // int(2048 * 0.05)
#define REP_H  0.0005f

// ---------------------------------------------------------------------------
// accumulator layout in ws (floats):
//   acc[0]    = chamfer weighted sum
//   acc[1]    = repulsion sum
//   acc[2..9] = per-batch NN sqrt sums (current percentage)
//   acc[10]   = uniform-loss accumulator
// ---------------------------------------------------------------------------

__global__ void k_init(float* acc) {
    if (threadIdx.x < 16) acc[threadIdx.x] = 0.0f;
}

__global__ void k_sumsq(const float* __restrict__ pts, float* __restrict__ ss) {
    int g = blockIdx.x * blockDim.x + threadIdx.x;
    if (g >= BATCH * NPTS) return;
    const float* p = pts + (size_t)g * 3;
    ss[g] = p[0]*p[0] + p[1]*p[1] + p[2]*p[2];
}

// ---------------------------------------------------------------------------
// Chamfer: for each row of X, min over all Y of d2 = |x|^2 + |y|^2 - 2 x.y
// Cross term computed with V_WMMA_F32_16X16X4_F32 (K=4, 4th component = 0).
// A 16x4 f32 layout: lanes 0-15 row M=lane, VGPR0=K0 VGPR1=K1;
//                    lanes 16-31 row M=lane-16, VGPR0=K2 VGPR1=K3.
// B 4x16 mirrored.  D: VGPR r -> lanes0-15 M=r, lanes16-31 M=r+8, N=lane&15.
// ---------------------------------------------------------------------------
__global__ void __launch_bounds__(256)
k_cd_rowmin(const float* __restrict__ X, const float* __restrict__ ssX,
            const float* __restrict__ Y, const float* __restrict__ ssY,
            float* __restrict__ acc, float weight) {
    const int wv   = threadIdx.x >> 5;
    const int lane = threadIdx.x & 31;
    const int tile = blockIdx.x * 8 + wv;            // 16-row tile id, 0..1023
    const int b       = tile / (NPTS / 16);
    const int rowBase = (tile % (NPTS / 16)) * 16;
    const int q  = lane & 15;
    const bool hi = lane >= 16;

    const float* Xb = X + (size_t)b * NPTS * 3;
    const float* Yb = Y + (size_t)b * NPTS * 3;

    const int m = rowBase + q;
    const float ax = Xb[m*3+0], ay = Xb[m*3+1], az = Xb[m*3+2];
    v2f A;
    A.x = hi ? az : ax;      // K=0 (lo half) / K=2 (hi half)
    A.y = hi ? 0.0f : ay;    // K=1 (lo half) / K=3 (hi half, padded 0)

    float rmin[8];
    #pragma unroll
    for (int r = 0; r < 8; ++r) rmin[r] = 3.0e38f;

    for (int ct = 0; ct < NPTS / 16; ++ct) {
        const int n = ct * 16 + q;
        const float bx = Yb[n*3+0], by = Yb[n*3+1], bz = Yb[n*3+2];
        v2f Bm;
        Bm.x = hi ? bz : bx;
        Bm.y = hi ? 0.0f : by;
        v8f c = {};
        c = __builtin_amdgcn_wmma_f32_16x16x4_f32(false, A, false, Bm,
                                                  (short)0, c, false, false);
        const float ssn = ssY[b * NPTS + n];
        #pragma unroll
        for (int r = 0; r < 8; ++r)
            rmin[r] = fminf(rmin[r], ssn - 2.0f * c[r]);
    }

    // min across each 16-lane half (all 16 columns of the tile set per half)
    #pragma unroll
    for (int r = 0; r < 8; ++r) {
        #pragma unroll
        for (int msk = 8; msk >= 1; msk >>= 1)
            rmin[r] = fminf(rmin[r], __shfl_xor(rmin[r], msk, 32));
    }

    if (q == 0) {            // lanes 0 and 16 hold the reduced halves
        float s = 0.0f;
        #pragma unroll
        for (int r = 0; r < 8; ++r) {
            const int row = rowBase + r + (hi ? 8 : 0);
            const float d = rmin[r] + ssX[b * NPTS + row];
            s += fmaxf(d, 0.0f);     // reference clamps d2 at 0
        }
        atomicAdd(acc, weight * s);
    }
}

// ---------------------------------------------------------------------------
// Repulsion: 4 nearest neighbors (excluding self, which is the 0-distance min)
// ---------------------------------------------------------------------------
__global__ void k_repulsion(const float* __restrict__ P, float* __restrict__ acc) {
    int g = blockIdx.x * blockDim.x + threadIdx.x;
    if (g >= BATCH * NPTS) return;
    const int b = g / NPTS, i = g % NPTS;
    const float* Pb = P + (size_t)b * NPTS * 3;
    const float px = Pb[i*3], py = Pb[i*3+1], pz = Pb[i*3+2];
    float best[5] = {3e38f, 3e38f, 3e38f, 3e38f, 3e38f};
    for (int j = 0; j < NPTS; ++j) {
        const float dx = Pb[j*3]-px, dy = Pb[j*3+1]-py, dz = Pb[j*3+2]-pz;
        const float d2 = dx*dx + dy*dy + dz*dz;
        if (d2 < best[4]) {
            best[4] = d2;
            #pragma unroll
            for (int k = 4; k > 0; --k)
                if (best[k] < best[k-1]) { float t = best[k]; best[k] = best[k-1]; best[k-1] = t; }
        }
    }
    float s = 0.0f;
    #pragma unroll
    for (int k = 1; k < 5; ++k) s += fmaxf(REP_H - best[k], 0.0f);
    atomicAdd(acc + 1, s);
}

// ---------------------------------------------------------------------------
// Farthest point sampling: one workgroup per batch, sequential scan steps,
// LDS argmax with first-occurrence tie-break (matches jnp.argmax).
// ---------------------------------------------------------------------------
__global__ void __launch_bounds__(256)
k_fps(const float* __restrict__ P, int* __restrict__ fidx) {
    __shared__ float sval[256];
    __shared__ int   sidx[256];
    const int b = blockIdx.x, tid = threadIdx.x;
    const float* Pb = P + (size_t)b * NPTS * 3;
    float dreg[NPTS / 256];
    #pragma unroll
    for (int k = 0; k < NPTS / 256; ++k) dreg[k] = 1e10f;
    int last = 0;
    if (tid == 0) fidx[b * NPOINT] = 0;
    for (int s = 1; s < NPOINT; ++s) {
        const float px = Pb[last*3], py = Pb[last*3+1], pz = Pb[last*3+2];
        float bv = -1.0f; int bi = 0;
        #pragma unroll
        for (int k = 0; k < NPTS / 256; ++k) {
            const int i = tid + k * 256;
            const float dx = Pb[i*3]-px, dy = Pb[i*3+1]-py, dz = Pb[i*3+2]-pz;
            const float nd = fminf(dreg[k], dx*dx + dy*dy + dz*dz);
            dreg[k] = nd;
            if (nd > bv) { bv = nd; bi = i; }   // strict >: earliest index wins
        }
        sval[tid] = bv; sidx[tid] = bi;
        __syncthreads();
        for (int off = 128; off > 0; off >>= 1) {
            if (tid < off) {
                const float v = sval[tid + off]; const int i2 = sidx[tid + off];
                if (v > sval[tid] || (v == sval[tid] && i2 < sidx[tid])) {
                    sval[tid] = v; sidx[tid] = i2;
                }
            }
            __syncthreads();
        }
        last = sidx[0];
        if (tid == 0) fidx[b * NPOINT + s] = last;
        __syncthreads();
    }
}

// ---------------------------------------------------------------------------
// Ball query: ascending scan == sorted candidate indices; pad with first hit.
// ---------------------------------------------------------------------------
__global__ void k_ballquery(const float* __restrict__ P, const int* __restrict__ fidx,
                            int* __restrict__ bq, int ns, float r2) {
    int g = blockIdx.x * blockDim.x + threadIdx.x;
    if (g >= BATCH * NPOINT) return;
    const int b = g / NPOINT, ci = g % NPOINT;
    const float* Pb = P + (size_t)b * NPTS * 3;
    const int c = fidx[b * NPOINT + ci];
    const float cx = Pb[c*3], cy = Pb[c*3+1], cz = Pb[c*3+2];
    int* o = bq + (size_t)g * ns;
    int cnt = 0, first = 0;
    bool got = false;
    for (int j = 0; j < NPTS; ++j) {
        const float dx = Pb[j*3]-cx, dy = Pb[j*3+1]-cy, dz = Pb[j*3+2]-cz;
        if (dx*dx + dy*dy + dz*dz < r2) {
            if (!got) { got = true; first = j; }
            if (cnt < ns) o[cnt++] = j;
        }
    }
    for (; cnt < ns; ++cnt) o[cnt] = first;
}

__global__ void k_gather(const float* __restrict__ P, const int* __restrict__ bq,
                         float* __restrict__ G, int M) {
    int g = blockIdx.x * blockDim.x + threadIdx.x;
    if (g >= BATCH * M) return;
    const int b = g / M, k = g % M;
    const int j = bq[(size_t)b * M + k];
    const float* s = P + ((size_t)b * NPTS + j) * 3;
    float* d = G + (size_t)g * 3;
    d[0] = s[0]; d[1] = s[1]; d[2] = s[2];
}

// nearest neighbor excluding self (self d2 = 0 is always the first NN)
__global__ void k_nn(const float* __restrict__ G, float* __restrict__ acc, int M) {
    int g = blockIdx.x * blockDim.x + threadIdx.x;
    if (g >= BATCH * M) return;
    const int b = g / M, i = g % M;
    const float* Gb = G + (size_t)b * M * 3;
    const float px = Gb[i*3], py = Gb[i*3+1], pz = Gb[i*3+2];
    float dmin = 3e38f;
    for (int j = 0; j < M; ++j) {
        if (j == i) continue;
        const float dx = Gb[j*3]-px, dy = Gb[j*3+1]-py, dz = Gb[j*3+2]-pz;
        dmin = fminf(dmin, dx*dx + dy*dy + dz*dz);
    }
    atomicAdd(acc + 2 + b, sqrtf(fmaxf(dmin, 0.0f)) + 1e-8f);
}

__global__ void k_fin_p(float* acc, int M, float expect_len, float pw) {
    float s = 0.0f;
    for (int b = 0; b < BATCH; ++b) {
        const float u = acc[2 + b] / (float)M;
        const float d = u - expect_len;
        s += d * d / (expect_len + 1e-8f);
        acc[2 + b] = 0.0f;          // reset for next percentage
    }
    acc[10] += pw * s / (float)BATCH;
}

__global__ void k_final(const float* acc, const float* rad, float* out) {
    const float cd  = acc[0] / (float)(BATCH * NPTS) / rad[0];
    const float rep = acc[1] / (float)(BATCH * NPTS * 4);
    const float uni = acc[10] / 5.0f;
    out[0] = cd + rep + uni;
}

// ---------------------------------------------------------------------------
extern "C" void kernel_launch(void* const* d_in, const int* in_sizes, int n_in,
                              void* d_out, int out_size, void* d_ws, size_t ws_size,
                              hipStream_t stream) {
    (void)in_sizes; (void)n_in; (void)out_size; (void)ws_size;
    const float* pred = (const float*)d_in[0];
    const float* gt   = (const float*)d_in[1];
    const float* rad  = (const float*)d_in[2];
    float* out = (float*)d_out;

    float* acc     = (float*)d_ws;                         // 16 floats
    float* ssp     = acc + 16;                             // BATCH*NPTS
    float* ssg     = ssp + BATCH * NPTS;                   // BATCH*NPTS
    int*   fidx    = (int*)(ssg + BATCH * NPTS);           // BATCH*NPOINT
    int*   bqidx   = fidx + BATCH * NPOINT;                // BATCH*NPOINT*24 max
    float* grouped = (float*)(bqidx + BATCH * NPOINT * 24);// BATCH*NPOINT*24*3

    k_init<<<1, 32, 0, stream>>>(acc);
    k_sumsq<<<(BATCH * NPTS + 255) / 256, 256, 0, stream>>>(pred, ssp);
    k_sumsq<<<(BATCH * NPTS + 255) / 256, 256, 0, stream>>>(gt, ssg);

    // Chamfer: cost_for (per-gt min over pred, w=0.8), cost_bac (w=0.2)
    k_cd_rowmin<<<BATCH * NPTS / 128, 256, 0, stream>>>(gt, ssg, pred, ssp, acc, 0.8f);
    k_cd_rowmin<<<BATCH * NPTS / 128, 256, 0, stream>>>(pred, ssp, gt, ssg, acc, 0.2f);

    k_repulsion<<<(BATCH * NPTS + 255) / 256, 256, 0, stream>>>(pred, acc);

    k_fps<<<BATCH, 256, 0, stream>>>(pred, fidx);

    const int   nsamples[5] = {8, 12, 16, 20, 24};  // int(2048*p)
    const float ps[5] = {0.004f, 0.006f, 0.008f, 0.01f, 0.012f};
    const float expect_len = sqrtf((float)(M_PI / 2048.0));
    for (int t = 0; t < 5; ++t) {
        const int   ns = nsamples[t];
        const int   M  = NPOINT * ns;
        const float r2 = ps[t];                      // r = sqrt(p*1) -> r^2 = p
        const float pw = (ps[t] * 100.0f) * (ps[t] * 100.0f);
        k_ballquery<<<(BATCH * NPOINT + 63) / 64, 64, 0, stream>>>(pred, fidx, bqidx, ns, r2);
        k_gather<<<(BATCH * M + 255) / 256, 256, 0, stream>>>(pred, bqidx, grouped, M);
        k_nn<<<(BATCH * M + 127) / 128, 128, 0, stream>>>(grouped, acc, M);
        k_fin_p<<<1, 1, 0, stream>>>(acc, M, expect_len, pw);
    }
    k_final<<<1, 1, 0, stream>>>(acc, rad, out);
}